// InstanceSegLoss_20160576487539
// MI455X (gfx1250) — compile-verified
//
#include <hip/hip_runtime.h>
#include <hip/hip_bf16.h>
#include <stdint.h>

// ---------------- problem constants (from reference) ----------------
#define B_   16
#define N_   65536
#define C_   20
#define E_   16
#define K_   64
#define DELTA_V 0.5f
#define DELTA_D 1.5f
#define GAMMA_W 0.001f

#define WAVE 32

typedef float v2f __attribute__((ext_vector_type(2)));
typedef float v8f __attribute__((ext_vector_type(8)));

// ---------------- workspace layout (floats) ----------------
#define OFF_SUMS   0                        // B*K*E = 16384
#define OFF_COUNTS (OFF_SUMS + B_*K_*E_)    // B*K   = 1024
#define OFF_DSUM   (OFF_COUNTS + B_*K_)     // B*K   = 1024
#define OFF_MEANS  (OFF_DSUM + B_*K_)       // B*K*E = 16384
#define OFF_NORM2  (OFF_MEANS + B_*K_*E_)   // B*K   = 1024
#define OFF_ACC    (OFF_NORM2 + B_*K_)      // [0]=sem [1]=cen [2]=ins (+pad)
#define WS_FLOATS  (OFF_ACC + 8)

// ---------------- kernel 0: zero accumulators ----------------
__global__ void k_zero(float* ws) {
  int i = blockIdx.x * blockDim.x + threadIdx.x;
  if (i < WS_FLOATS) ws[i] = 0.0f;
}

// ---------------- kernel 1: semantic CE + center MSE ----------------
__global__ void k_sem_center(const float* __restrict__ logits,
                             const int*   __restrict__ slab,
                             const float* __restrict__ cpred,
                             const float* __restrict__ clab,
                             float* __restrict__ acc) {
  const int64_t total = (int64_t)B_ * N_;
  float sem = 0.f, cen = 0.f;
  for (int64_t idx = (int64_t)blockIdx.x * blockDim.x + threadIdx.x; idx < total;
       idx += (int64_t)gridDim.x * blockDim.x) {
    int b = (int)(idx >> 16);       // N_ == 65536
    int n = (int)(idx & 65535);
    const float* lp = logits + ((int64_t)b * C_) * N_ + n;
    float v[C_];
    float mx = -3.402823466e+38f;
#pragma unroll
    for (int c = 0; c < C_; ++c) { v[c] = lp[(int64_t)c * N_]; mx = fmaxf(mx, v[c]); }
    float se = 0.f;
#pragma unroll
    for (int c = 0; c < C_; ++c) se += __expf(v[c] - mx);
    int lab = slab[idx];
    sem += (mx + __logf(se)) - v[lab];

    const float* cp = cpred + idx * 3;
    const float* cl = clab  + idx * 3;
#pragma unroll
    for (int j = 0; j < 3; ++j) { float d = cp[j] - cl[j]; cen += d * d; }
  }
#pragma unroll
  for (int off = WAVE / 2; off > 0; off >>= 1) {
    sem += __shfl_down(sem, off, WAVE);
    cen += __shfl_down(cen, off, WAVE);
  }
  if ((threadIdx.x & (WAVE - 1)) == 0) {
    atomicAdd(&acc[0], sem);
    atomicAdd(&acc[1], cen);
  }
}

// ---------------- kernel 2: per-instance segment sums of embeddings ----------------
// Dim-parallel layout: 16-lane group owns one point, lane e owns dim e.
// Loads are fully coalesced; per-point LDS atomics hit 16 distinct banks in parallel.
#define SEG_PTS 4096
__global__ void k_segsum(const float* __restrict__ emb, const int* __restrict__ ilab,
                         float* __restrict__ sums, float* __restrict__ counts) {
  __shared__ float lacc[K_ * E_];
  __shared__ float lcnt[K_];
  const int blocksPerBatch = N_ / SEG_PTS;
  int b = blockIdx.x / blocksPerBatch;
  int chunk = blockIdx.x % blocksPerBatch;
  for (int i = threadIdx.x; i < K_ * E_; i += blockDim.x) lacc[i] = 0.f;
  for (int i = threadIdx.x; i < K_;      i += blockDim.x) lcnt[i] = 0.f;
  __syncthreads();
  int g = threadIdx.x >> 4;        // 0..15 point-group within block
  int e = threadIdx.x & 15;        // dim
  int64_t base = (int64_t)b * N_ + (int64_t)chunk * SEG_PTS;
  for (int i = 0; i < SEG_PTS / 16; ++i) {
    int64_t p = base + (int64_t)i * 16 + g;
    int lab = ilab[p];
    float v = emb[p * E_ + e];
    atomicAdd(&lacc[lab * E_ + e], v);
    if (e == 0) atomicAdd(&lcnt[lab], 1.f);
  }
  __syncthreads();
  float* gs = sums + (int64_t)b * K_ * E_;
  for (int i = threadIdx.x; i < K_ * E_; i += blockDim.x) atomicAdd(&gs[i], lacc[i]);
  for (int i = threadIdx.x; i < K_;      i += blockDim.x) atomicAdd(&counts[b * K_ + i], lcnt[i]);
}

// ---------------- kernel 3: means and squared norms ----------------
__global__ void k_means(const float* __restrict__ sums, const float* __restrict__ counts,
                        float* __restrict__ means, float* __restrict__ norms2) {
  int b = blockIdx.x;
  int k = threadIdx.x;                 // 64 threads
  float cnt = counts[b * K_ + k];
  float inv = 1.f / fmaxf(cnt, 1.f);
  float n2 = 0.f;
#pragma unroll
  for (int e = 0; e < E_; ++e) {
    float m = sums[(b * K_ + k) * E_ + e] * inv;
    means[(b * K_ + k) * E_ + e] = m;
    n2 += m * m;
  }
  norms2[b * K_ + k] = n2;
}

// ---------------- kernel 4: variance term segment sums ----------------
// Same dim-parallel layout; group-of-16 shfl_xor reduction for the 16-dim sqdist.
__global__ void k_var(const float* __restrict__ emb, const int* __restrict__ ilab,
                      const float* __restrict__ means, float* __restrict__ dsum) {
  __shared__ float lm[K_ * E_];
  __shared__ float ld[K_];
  const int blocksPerBatch = N_ / SEG_PTS;
  int b = blockIdx.x / blocksPerBatch;
  int chunk = blockIdx.x % blocksPerBatch;
  for (int i = threadIdx.x; i < K_ * E_; i += blockDim.x) lm[i] = means[(int64_t)b * K_ * E_ + i];
  for (int i = threadIdx.x; i < K_;      i += blockDim.x) ld[i] = 0.f;
  __syncthreads();
  int g = threadIdx.x >> 4;
  int e = threadIdx.x & 15;
  int64_t base = (int64_t)b * N_ + (int64_t)chunk * SEG_PTS;
  for (int i = 0; i < SEG_PTS / 16; ++i) {
    int64_t p = base + (int64_t)i * 16 + g;
    int lab = ilab[p];
    float d1 = emb[p * E_ + e] - lm[lab * E_ + e];
    float sq = d1 * d1;
    // reduce across the 16-lane group (stays within wave32 half)
    sq += __shfl_xor(sq, 1, WAVE);
    sq += __shfl_xor(sq, 2, WAVE);
    sq += __shfl_xor(sq, 4, WAVE);
    sq += __shfl_xor(sq, 8, WAVE);
    if (e == 0) {
      float d = sq > 0.f ? sqrtf(sq) : 0.f;    // _safe_norm semantics
      d = fmaxf(d - DELTA_V, 0.f);
      atomicAdd(&ld[lab], d);
    }
  }
  __syncthreads();
  for (int i = threadIdx.x; i < K_; i += blockDim.x) atomicAdd(&dsum[b * K_ + i], ld[i]);
}

// ---------------- kernel 5: per-batch finalize (WMMA f32 16x16x4 Gram) ----------------
__global__ void k_final(const float* __restrict__ means, const float* __restrict__ norms2,
                        const float* __restrict__ counts, const float* __restrict__ dsum,
                        float* __restrict__ acc) {
  __shared__ float lm[K_ * E_];
  __shared__ float ln2[K_];
  __shared__ float lpres[K_];
  int b = blockIdx.x;
  int lane = threadIdx.x;               // 32 lanes, one wave, EXEC all ones
  for (int i = lane; i < K_ * E_; i += WAVE) lm[i] = means[(int64_t)b * K_ * E_ + i];
  for (int i = lane; i < K_;      i += WAVE) ln2[i] = norms2[b * K_ + i];
  __syncthreads();

  // per-instance stats: lane handles k = lane and k = lane + 32
  float var = 0.f, reg = 0.f, Mf = 0.f;
#pragma unroll
  for (int h = 0; h < 2; ++h) {
    int k = lane + h * WAVE;
    float cnt = counts[b * K_ + k];
    float pres = cnt > 0.f ? 1.f : 0.f;
    lpres[k] = pres;
    var += pres * (dsum[b * K_ + k] / fmaxf(cnt, 1.f));
    float n2 = ln2[k];
    reg += pres * (n2 > 0.f ? sqrtf(n2) : 0.f);
    Mf += pres;
  }
  __syncthreads();

  // Gram matrix of means via V_WMMA_F32_16X16X4_F32 (exact fp32)
  // A 16x4 layout: lanes 0-15 hold K=0,1 ; lanes 16-31 hold K=2,3
  float pen = 0.f;
  int m = lane & 15;
  int k0 = (lane >> 4) * 2;
  int ihalf = (lane >> 4) * 8;
  for (int it = 0; it < 4; ++it) {
    for (int jt = it; jt < 4; ++jt) {
      v8f cfrag = {};
#pragma unroll
      for (int kc = 0; kc < 4; ++kc) {
        v2f a, bf;
        a.x  = lm[(it * 16 + m) * E_ + kc * 4 + k0];
        a.y  = lm[(it * 16 + m) * E_ + kc * 4 + k0 + 1];
        bf.x = lm[(jt * 16 + m) * E_ + kc * 4 + k0];   // B = means^T tile
        bf.y = lm[(jt * 16 + m) * E_ + kc * 4 + k0 + 1];
        cfrag = __builtin_amdgcn_wmma_f32_16x16x4_f32(
            false, a, false, bf, (short)0, cfrag, false, false);
      }
      // C/D layout: VGPR vg -> row M = vg (+8 for lanes 16-31), col N = lane&15
      int j = jt * 16 + m;
      float presj = lpres[j];
      float nj = ln2[j];
#pragma unroll
      for (int vg = 0; vg < 8; ++vg) {
        int i = it * 16 + ihalf + vg;
        float presi = lpres[i];          // unconditional loads -> no exec branching
        float ni = ln2[i];
        float sq = ni + nj - 2.f * cfrag[vg];
        float D = sq > 0.f ? sqrtf(sq) : 0.f;
        float p = fmaxf(2.f * DELTA_D - D, 0.f);
        float ltf = (i < j) ? 1.0f : 0.0f;   // pure register select (v_cndmask)
        pen = fmaf(presi * presj * ltf, p, pen);
      }
    }
  }

#pragma unroll
  for (int off = WAVE / 2; off > 0; off >>= 1) {
    pen += __shfl_down(pen, off, WAVE);
    var += __shfl_down(var, off, WAVE);
    reg += __shfl_down(reg, off, WAVE);
    Mf  += __shfl_down(Mf,  off, WAVE);
  }
  if (lane == 0) {
    // lower-tri + diagonal entries each contribute 2*delta_d (masked-before-clamp quirk)
    float sum_pen = pen + (Mf * (Mf + 1.f) * 0.5f) * (2.f * DELTA_D);
    float dist = (Mf > 1.f) ? sum_pen / fmaxf(Mf * (Mf - 1.f), 1.f) : 0.f;
    float ok   = (Mf > 1.f) ? 1.f : 0.f;
    float regv = ok * (reg / fmaxf(Mf, 1.f));
    float varv = ok * var;
    float contrib = (varv + dist + GAMMA_W * regv) * (1.f / (float)B_);
    atomicAdd(&acc[2], contrib);
  }
}

// ---------------- kernel 6: combine scalars ----------------
__global__ void k_finish(const float* __restrict__ acc, float* __restrict__ out) {
  float sem = acc[0] / (float)((int64_t)B_ * N_);
  float cen = acc[1] / (float)((int64_t)B_ * N_ * 3);
  out[0] = sem + cen + acc[2];
}

// ---------------- launch ----------------
extern "C" void kernel_launch(void* const* d_in, const int* in_sizes, int n_in,
                              void* d_out, int out_size, void* d_ws, size_t ws_size,
                              hipStream_t stream) {
  (void)in_sizes; (void)n_in; (void)out_size; (void)ws_size;
  const float* logits = (const float*)d_in[0];
  const int*   slab   = (const int*)d_in[1];
  const float* cpred  = (const float*)d_in[2];
  const float* clab   = (const float*)d_in[3];
  const float* emb    = (const float*)d_in[4];
  const int*   ilab   = (const int*)d_in[5];
  float* ws  = (float*)d_ws;
  float* out = (float*)d_out;

  float* sums   = ws + OFF_SUMS;
  float* counts = ws + OFF_COUNTS;
  float* dsum   = ws + OFF_DSUM;
  float* means  = ws + OFF_MEANS;
  float* norms2 = ws + OFF_NORM2;
  float* acc    = ws + OFF_ACC;

  k_zero<<<(WS_FLOATS + 255) / 256, 256, 0, stream>>>(ws);
  k_sem_center<<<4096, 256, 0, stream>>>(logits, slab, cpred, clab, acc);
  k_segsum<<<B_ * (N_ / SEG_PTS), 256, 0, stream>>>(emb, ilab, sums, counts);
  k_means<<<B_, K_, 0, stream>>>(sums, counts, means, norms2);
  k_var<<<B_ * (N_ / SEG_PTS), 256, 0, stream>>>(emb, ilab, means, dsum);
  k_final<<<B_, WAVE, 0, stream>>>(means, norms2, counts, dsum, acc);
  k_finish<<<1, 1, 0, stream>>>(acc, out);
}